// NearestNbrNegSampler_81174881894680
// MI455X (gfx1250) — compile-verified
//
#include <hip/hip_runtime.h>

#define AS1 __attribute__((address_space(1)))
#define AS3 __attribute__((address_space(3)))

typedef float v4f __attribute__((ext_vector_type(4)));

// One block (256 threads, 8 wave32s) per batch row.
// Phase A: stage y/l/u row into LDS (async-to-LDS on gfx1250), compute exact
//          per-neighbor masks via ballot-based violation counting.
// Phase B: stream the 512x256 fp32 neighbor tile with nontemporal B128 stores.
__global__ __launch_bounds__(256) void nnns_kernel(
    const float* __restrict__ l,
    const float* __restrict__ u,
    const float* __restrict__ y,
    float* __restrict__ n_out,
    float* __restrict__ m_out)
{
    constexpr int D = 256;
    const int row = blockIdx.x;
    const int t   = threadIdx.x;

    __shared__ __align__(16) float sy[D];
    __shared__ __align__(16) float sl[D];
    __shared__ __align__(16) float su[D];
    __shared__ unsigned s_viol[8];

    const float* gy = y + (size_t)row * D;
    const float* gl = l + (size_t)row * D;
    const float* gu = u + (size_t)row * D;

#if __has_builtin(__builtin_amdgcn_global_load_async_to_lds_b32)
    // CDNA5 async global->LDS path (ASYNCcnt tracked). One B32 per lane per row.
    // Builtin signature (from hipcc diagnostic): (AS1 int*, AS3 int*, imm, imm)
    __builtin_amdgcn_global_load_async_to_lds_b32(
        (AS1 int*)(gy + t), (AS3 int*)(sy + t), 0, 0);
    __builtin_amdgcn_global_load_async_to_lds_b32(
        (AS1 int*)(gl + t), (AS3 int*)(sl + t), 0, 0);
    __builtin_amdgcn_global_load_async_to_lds_b32(
        (AS1 int*)(gu + t), (AS3 int*)(su + t), 0, 0);
#if __has_builtin(__builtin_amdgcn_s_wait_asynccnt)
    __builtin_amdgcn_s_wait_asynccnt(0);
#else
    asm volatile("s_wait_asynccnt 0" ::: "memory");
#endif
#else
    sy[t] = gy[t];
    sl[t] = gl[t];
    su[t] = gu[t];
#endif
    __syncthreads();

    // ---- Phase A: masks -------------------------------------------------
    const float yv = sy[t];
    const float lv = sl[t];
    const float uv = su[t];
    const bool in_range = (lv <= yv) && (yv <= uv);

    // wave32 ballot -> per-wave violation count -> block total via LDS
    unsigned long long bal = __ballot(!in_range);
    unsigned wave_viol = (unsigned)__popcll(bal);
    if ((t & 31) == 0) s_viol[t >> 5] = wave_viol;
    __syncthreads();
    unsigned viol = 0;
#pragma unroll
    for (int w = 0; w < 8; ++w) viol += s_viol[w];

    // violations among elements k != t
    const unsigned others = viol - (in_range ? 0u : 1u);
    const float yp = yv + 1.0f;
    const float ym = yv - 1.0f;
    const bool mask_plus  = (others == 0u) && (lv <= yp) && (yp <= uv);
    const bool mask_minus = (others == 0u) && (lv <= ym) && (ym <= uv);

    float* mrow = m_out + (size_t)row * (2 * D);
    mrow[t]     = mask_plus  ? 1.0f : 0.0f;
    mrow[D + t] = mask_minus ? 1.0f : 0.0f;

    // ---- Phase B: stream n = y +/- e_j ---------------------------------
    // 512 rows x 64 float4 = 32768 float4 per block; 128 iters x 256 threads.
    const v4f* sy4 = (const v4f*)sy;           // LDS, ds_load_b128
    float* nb = n_out + (size_t)row * (2 * D) * D;

#pragma unroll 4
    for (int it = 0; it < 128; ++it) {
        const int q  = it * 256 + t;           // float4 linear index
        const int j  = q >> 6;                 // neighbor row 0..511
        const int k4 = q & 63;                 // float4 within row

        v4f v = sy4[k4];
        const int   jj  = j & 255;             // which basis vector
        const float sgn = (j < 256) ? 1.0f : -1.0f;
        const int   c   = jj - (k4 << 2);      // component hit by e_jj
        if ((unsigned)c < 4u) v[c] += sgn;

        __builtin_nontemporal_store(v, (v4f*)(nb + (size_t)q * 4));
    }
}

extern "C" void kernel_launch(void* const* d_in, const int* in_sizes, int n_in,
                              void* d_out, int out_size, void* d_ws, size_t ws_size,
                              hipStream_t stream) {
    // setup_inputs order: a(0) b(1) c(2) l(3) u(4) h(5) y(6)
    const float* l = (const float*)d_in[3];
    const float* u = (const float*)d_in[4];
    const float* y = (const float*)d_in[6];

    const int D = 256;
    const int B = in_sizes[6] / D;             // 2048

    float* n_out = (float*)d_out;                              // B*2D*D floats
    float* m_out = n_out + (size_t)B * (2 * D) * D;            // B*2D mask floats

    nnns_kernel<<<B, 256, 0, stream>>>(l, u, y, n_out, m_out);
}